// GATEncoder_51402168598555
// MI455X (gfx1250) — compile-verified
//
#include <hip/hip_runtime.h>
#include <cstdint>
#include <cstddef>

typedef float v2f __attribute__((ext_vector_type(2)));
typedef float v8f __attribute__((ext_vector_type(8)));

#define IN_DIM 128
#define HID 64
#define HEADS 4

// ---------------------------------------------------------------------------
// CSR construction over destination nodes (self-loop stored first per node)
// ---------------------------------------------------------------------------
__global__ void k_init_deg(int* __restrict__ deg, int n) {
  int i = blockIdx.x * blockDim.x + threadIdx.x;
  if (i < n) deg[i] = 1;  // self-loop
}

__global__ void k_count(const int* __restrict__ dst, int E, int* __restrict__ deg) {
  int e = blockIdx.x * blockDim.x + threadIdx.x;
  if (e < E) atomicAdd(&deg[dst[e]], 1);
}

// exclusive scan of deg[0..n) -> row_off[0..n], single 1024-thread block
__global__ void k_scan(const int* __restrict__ deg, int* __restrict__ row_off, int n) {
  __shared__ int partial[1024];
  const int t = threadIdx.x;
  const int per = (n + 1023) / 1024;
  const int start = t * per;
  int local = 0;
  for (int i = 0; i < per; ++i) {
    int idx = start + i;
    if (idx < n) local += deg[idx];
  }
  partial[t] = local;
  __syncthreads();
  for (int off = 1; off < 1024; off <<= 1) {
    int v = (t >= off) ? partial[t - off] : 0;
    __syncthreads();
    partial[t] += v;
    __syncthreads();
  }
  int run = (t == 0) ? 0 : partial[t - 1];
  for (int i = 0; i < per; ++i) {
    int idx = start + i;
    if (idx < n) { row_off[idx] = run; run += deg[idx]; }
  }
  if (t == 1023) row_off[n] = partial[1023];
}

__global__ void k_selfloop(const int* __restrict__ row_off, int* __restrict__ col,
                           int* __restrict__ cursor, int n) {
  int i = blockIdx.x * blockDim.x + threadIdx.x;
  if (i < n) { int p = row_off[i]; col[p] = i; cursor[i] = p + 1; }
}

__global__ void k_fill(const int* __restrict__ src, const int* __restrict__ dst, int E,
                       int* __restrict__ cursor, int* __restrict__ col) {
  int e = blockIdx.x * blockDim.x + threadIdx.x;
  if (e < E) { int p = atomicAdd(&cursor[dst[e]], 1); col[p] = src[e]; }
}

// ---------------------------------------------------------------------------
// fp32 GEMM: D[M,N] = A[M,K] * B[K,N] using V_WMMA_F32_16X16X4_F32.
// One block per 16-row stripe of A (staged in LDS, padded), one wave per
// 16-column tile. blockDim.x = (N/16)*32.
// ---------------------------------------------------------------------------
__global__ void k_gemm_wmma(const float* __restrict__ A, const float* __restrict__ B,
                            float* __restrict__ D, int M, int K, int N) {
  extern __shared__ float lds[];  // 16 x (K+4)
  const int row0 = blockIdx.x * 16;
  const int wave = threadIdx.x >> 5;
  const int lane = threadIdx.x & 31;
  const int col0 = wave * 16;
  const int ldp = K + 4;  // pad to avoid 64-bank conflicts on column reads

  for (int i = threadIdx.x; i < 16 * K; i += blockDim.x) {
    int r = i / K, c = i - r * K;
    int gr = row0 + r;
    lds[r * ldp + c] = (gr < M) ? A[(size_t)gr * K + c] : 0.0f;
  }
  __syncthreads();

  // A 16x4 frag: lanes 0-15 -> K={k0,k0+1}, lanes 16-31 -> K={k0+2,k0+3}
  // B 4x16 frag: same K-grouping by lane half, lane&15 = column
  const int half = lane >> 4;
  const int idx = lane & 15;
  v8f acc = {};
#pragma unroll 4
  for (int k0 = 0; k0 < K; k0 += 4) {
    const int ka = k0 + half * 2;
    v2f a, b;
    a.x = lds[idx * ldp + ka];
    a.y = lds[idx * ldp + ka + 1];
    const float* bp = B + (size_t)ka * N + col0 + idx;
    b.x = bp[0];
    b.y = bp[N];
    acc = __builtin_amdgcn_wmma_f32_16x16x4_f32(false, a, false, b, (short)0, acc,
                                                false, false);
  }
  // D layout: lane&15 = n, VGPR v -> m = v + 8*(lane>>4)
#pragma unroll
  for (int v = 0; v < 8; ++v) {
    int m = row0 + v + 8 * half;
    if (m < M) D[(size_t)m * N + col0 + idx] = acc[v];
  }
}

// ---------------------------------------------------------------------------
// alpha_s/alpha_d: per (node, head) dot of h[n, head, :] with a_src/a_dst
// ---------------------------------------------------------------------------
__global__ void k_alpha(const float* __restrict__ h, const float* __restrict__ a_src,
                        const float* __restrict__ a_dst, float* __restrict__ as_out,
                        float* __restrict__ ad_out, int n, int H, int C) {
  int i = blockIdx.x * blockDim.x + threadIdx.x;
  if (i >= n * H) return;
  int node = i / H, hd = i - node * H;
  const float* hp = h + (size_t)node * H * C + (size_t)hd * C;
  const float* sp = a_src + (size_t)hd * C;
  const float* dp = a_dst + (size_t)hd * C;
  float accs = 0.0f, accd = 0.0f;
  for (int c = 0; c < C; c += 4) {
    float4 hv = *(const float4*)(hp + c);
    float4 sv = *(const float4*)(sp + c);
    float4 dv = *(const float4*)(dp + c);
    accs += hv.x * sv.x + hv.y * sv.y + hv.z * sv.z + hv.w * sv.w;
    accd += hv.x * dv.x + hv.y * dv.y + hv.z * dv.z + hv.w * dv.w;
  }
  as_out[i] = accs;
  ad_out[i] = accd;
}

// ---------------------------------------------------------------------------
// GAT aggregation: one wave32 per destination node.
// Per-node segment softmax (max + sum-exp via shfl_xor within head groups),
// then weighted gather of h[src] with each lane owning H*C/32 channels.
// Fuses bias add and optional ELU.
// ---------------------------------------------------------------------------
template <int H, int C>
__global__ void k_aggregate(const float* __restrict__ h, const float* __restrict__ as,
                            const float* __restrict__ ad, const int* __restrict__ row_off,
                            const int* __restrict__ col, const float* __restrict__ bias,
                            float* __restrict__ out, int n, int applyElu) {
  constexpr int HC = H * C;
  constexpr int CH = HC / 32;   // channels per lane (8 or 2)
  constexpr int LPH = 32 / H;   // lanes per head (8 or 32)
  const int wv = (int)((blockIdx.x * blockDim.x + threadIdx.x) >> 5);
  const int lane = threadIdx.x & 31;
  if (wv >= n) return;
  const int node = wv;
  const int ch0 = lane * CH;
  const int head = ch0 / C;     // each lane's channels lie in one head
  const int li = lane & (LPH - 1);
  const int beg = row_off[node], end = row_off[node + 1];
  const float adn = ad[node * H + head];

  // pass 1: per-head max of leaky_relu(alpha_s[src] + alpha_d[node])
  float mx = -3.402823466e+38f;
  for (int j = beg + li; j < end; j += LPH) {
    int s = col[j];
    float e = as[s * H + head] + adn;
    e = (e > 0.0f) ? e : 0.2f * e;
    mx = fmaxf(mx, e);
  }
#pragma unroll
  for (int m = LPH >> 1; m; m >>= 1) mx = fmaxf(mx, __shfl_xor(mx, m, 32));

  // pass 2: per-head sum of exp(e - max)
  float se = 0.0f;
  for (int j = beg + li; j < end; j += LPH) {
    int s = col[j];
    float e = as[s * H + head] + adn;
    e = (e > 0.0f) ? e : 0.2f * e;
    se += __expf(e - mx);
  }
#pragma unroll
  for (int m = LPH >> 1; m; m >>= 1) se += __shfl_xor(se, m, 32);
  const float inv = 1.0f / (se + 1e-16f);

  // pass 3: weighted gather-accumulate of h[src]
  float acc[CH];
#pragma unroll
  for (int i = 0; i < CH; ++i) acc[i] = 0.0f;
  for (int j = beg; j < end; ++j) {
    int s = col[j];
    float e = as[s * H + head] + adn;
    e = (e > 0.0f) ? e : 0.2f * e;
    float w = __expf(e - mx) * inv;
    const float* hp = h + (size_t)s * HC + ch0;
    if constexpr ((CH & 3) == 0) {
#pragma unroll
      for (int i = 0; i < CH; i += 4) {
        float4 v = *(const float4*)(hp + i);
        acc[i + 0] += w * v.x;
        acc[i + 1] += w * v.y;
        acc[i + 2] += w * v.z;
        acc[i + 3] += w * v.w;
      }
    } else {
#pragma unroll
      for (int i = 0; i < CH; ++i) acc[i] += w * hp[i];
    }
  }

  float* op = out + (size_t)node * HC + ch0;
#pragma unroll
  for (int i = 0; i < CH; ++i) {
    float v = acc[i] + bias[ch0 + i];
    if (applyElu) v = (v > 0.0f) ? v : (__expf(v) - 1.0f);
    op[i] = v;
  }
}

// ---------------------------------------------------------------------------
extern "C" void kernel_launch(void* const* d_in, const int* in_sizes, int n_in,
                              void* d_out, int out_size, void* d_ws, size_t ws_size,
                              hipStream_t stream) {
  const float* x   = (const float*)d_in[0];
  const int* edge  = (const int*)d_in[1];
  const float* W0  = (const float*)d_in[2];
  const float* as0 = (const float*)d_in[3];
  const float* ad0 = (const float*)d_in[4];
  const float* b0  = (const float*)d_in[5];
  const float* W1  = (const float*)d_in[6];
  const float* as1 = (const float*)d_in[7];
  const float* ad1 = (const float*)d_in[8];
  const float* b1  = (const float*)d_in[9];
  const float* W2  = (const float*)d_in[10];
  const float* as2 = (const float*)d_in[11];
  const float* ad2 = (const float*)d_in[12];
  const float* b2  = (const float*)d_in[13];

  const int N = in_sizes[0] / IN_DIM;  // 50000
  const int E = in_sizes[1] / 2;       // 800000
  const int Etot = E + N;              // edges + self-loops
  const int HC = HEADS * HID;          // 256

  const int* esrc = edge;
  const int* edst = edge + E;

  // carve workspace
  char* p = (char*)d_ws;
  auto carve = [&](size_t bytes) {
    char* r = p;
    p += (bytes + 255) & ~(size_t)255;
    return r;
  };
  float* hbuf    = (float*)carve((size_t)N * HC * sizeof(float));   // 51.2 MB
  float* gbuf    = (float*)carve((size_t)N * HC * sizeof(float));   // 51.2 MB
  float* asb     = (float*)carve((size_t)N * HEADS * sizeof(float));
  float* adb     = (float*)carve((size_t)N * HEADS * sizeof(float));
  int*   deg     = (int*)carve((size_t)N * sizeof(int));
  int*   cursor  = (int*)carve((size_t)N * sizeof(int));
  int*   row_off = (int*)carve((size_t)(N + 1) * sizeof(int));
  int*   col     = (int*)carve((size_t)Etot * sizeof(int));
  (void)ws_size; (void)n_in; (void)out_size;

  // ---- build dst-CSR (reused by all 3 layers) ----
  k_init_deg<<<(N + 255) / 256, 256, 0, stream>>>(deg, N);
  k_count<<<(E + 255) / 256, 256, 0, stream>>>(edst, E, deg);
  k_scan<<<1, 1024, 0, stream>>>(deg, row_off, N);
  k_selfloop<<<(N + 255) / 256, 256, 0, stream>>>(row_off, col, cursor, N);
  k_fill<<<(E + 255) / 256, 256, 0, stream>>>(esrc, edst, E, cursor, col);

  const int rowBlocks = (N + 15) / 16;
  const int aggBlocks = (N + 7) / 8;  // 8 waves (nodes) per 256-thread block

  // ---- layer 0: 128 -> 4x64 concat, ELU ----
  k_gemm_wmma<<<rowBlocks, (HC / 16) * 32, 16 * (IN_DIM + 4) * sizeof(float), stream>>>(
      x, W0, hbuf, N, IN_DIM, HC);
  k_alpha<<<(N * HEADS + 255) / 256, 256, 0, stream>>>(hbuf, as0, ad0, asb, adb, N, HEADS, HID);
  k_aggregate<HEADS, HID><<<aggBlocks, 256, 0, stream>>>(hbuf, asb, adb, row_off, col, b0,
                                                         gbuf, N, 1);

  // ---- layer 1: 256 -> 4x64 concat, ELU ----
  k_gemm_wmma<<<rowBlocks, (HC / 16) * 32, 16 * (HC + 4) * sizeof(float), stream>>>(
      gbuf, W1, hbuf, N, HC, HC);
  k_alpha<<<(N * HEADS + 255) / 256, 256, 0, stream>>>(hbuf, as1, ad1, asb, adb, N, HEADS, HID);
  k_aggregate<HEADS, HID><<<aggBlocks, 256, 0, stream>>>(hbuf, asb, adb, row_off, col, b1,
                                                         gbuf, N, 1);

  // ---- layer 2: 256 -> 64, single head, no ELU, writes d_out ----
  k_gemm_wmma<<<rowBlocks, (HID / 16) * 32, 16 * (HC + 4) * sizeof(float), stream>>>(
      gbuf, W2, hbuf, N, HC, HID);
  k_alpha<<<(N + 255) / 256, 256, 0, stream>>>(hbuf, as2, ad2, asb, adb, N, 1, HID);
  k_aggregate<1, HID><<<aggBlocks, 256, 0, stream>>>(hbuf, asb, adb, row_off, col, b2,
                                                     (float*)d_out, N, 0);
}